// GNN_stack_28750511079531
// MI455X (gfx1250) — compile-verified
//
#include <hip/hip_runtime.h>
#include <hip/hip_bf16.h>
#include <math.h>

#define N_NODES 100000
#define N_EDGES 1600000
#define D       128
#define D_OUT   40
#define AP      132   // LDS pitch for 128-wide activation rows (float4-aligned, bank stride 4)
#define WP      36    // LDS pitch for 32-wide weight slabs   (float4-aligned, bank stride 36)

typedef __attribute__((ext_vector_type(2))) float v2f;
typedef __attribute__((ext_vector_type(8))) float v8f;

#ifndef __has_builtin
#define __has_builtin(x) 0
#endif
#if __has_builtin(__builtin_amdgcn_global_load_async_to_lds_b128) && \
    __has_builtin(__builtin_amdgcn_s_wait_asynccnt)
#define USE_ASYNC_LDS 1
#else
#define USE_ASYNC_LDS 0
#endif

#if USE_ASYNC_LDS
typedef int cp_v4i __attribute__((vector_size(4 * sizeof(int))));
typedef __attribute__((address_space(1))) cp_v4i* cp_gp;  // global (AS1) int4*
typedef __attribute__((address_space(3))) cp_v4i* cp_lp;  // LDS (AS3) int4*
#endif

// copy 16B global -> LDS (per-lane). Async (ASYNCcnt) when available.
__device__ __forceinline__ void cp16(float* lds, const float* g) {
#if USE_ASYNC_LDS
    __builtin_amdgcn_global_load_async_to_lds_b128((cp_gp)g, (cp_lp)lds, 0, 0);
#else
    *(float4*)lds = *(const float4*)g;
#endif
}
__device__ __forceinline__ void cp_wait() {
#if USE_ASYNC_LDS
    __builtin_amdgcn_s_wait_asynccnt(0);
#endif
}

// ---------------------------------------------------------------------------
// zero workspace buffer (float4 granularity, exact launch)
// ---------------------------------------------------------------------------
__global__ void zero_kernel(float4* __restrict__ p, int n4) {
    int i = blockIdx.x * blockDim.x + threadIdx.x;
    if (i < n4) p[i] = make_float4(0.f, 0.f, 0.f, 0.f);
}

// ---------------------------------------------------------------------------
// scatter-add: prop[dst] += h[src], one wave (32 lanes) per edge, 4 floats/lane
// h and prop are L2-resident (51MB each, 192MB L2)
// ---------------------------------------------------------------------------
__global__ void scatter_kernel(const float* __restrict__ H,
                               const long long* __restrict__ ei,
                               float* __restrict__ P) {
    int e = blockIdx.x * 8 + (threadIdx.x >> 5);
    int lane = threadIdx.x & 31;
    long long s = ei[e];
    long long d = ei[N_EDGES + e];
    const float4 v = *(const float4*)&H[(size_t)s * D + lane * 4];
    float* p = &P[(size_t)d * D + lane * 4];
    atomicAdd(p + 0, v.x);
    atomicAdd(p + 1, v.y);
    atomicAdd(p + 2, v.z);
    atomicAdd(p + 3, v.w);
}

// ---------------------------------------------------------------------------
// Fused dual-GEMM SAGE layer:
//   out = relu( l2norm_rows( h@Wl^T + bl [+ prop@Wr^T + br] ) )   (DO_NORM=1)
//   out = h@Wl^T + bl                                             (DO_NORM=0, post-MLP1)
// 256 threads = 8 waves; each wave owns one 16x16 tile of the 16x128 output.
// f32 WMMA 16x16x4, K staged through LDS in 32-wide slabs.
// Two independent accumulator chains for WMMA issue ILP.
// ---------------------------------------------------------------------------
template <int HAS_PROP, int DO_NORM>
__global__ __launch_bounds__(256) void sage_gemm_kernel(
    const float* __restrict__ H, const float* __restrict__ P,
    const float* __restrict__ Wl, const float* __restrict__ bl,
    const float* __restrict__ Wr, const float* __restrict__ br,
    float* __restrict__ out) {
    __shared__ float sA[16 * AP];
    __shared__ float sP[16 * AP];
    __shared__ float sWl[128 * WP];
    __shared__ float sWr[128 * WP];
    __shared__ float rowsq[16];

    const int t = threadIdx.x;
    const int lane = t & 31;
    const int wave = t >> 5;           // 0..7 -> output columns wave*16..+15
    const int arow = lane & 15;        // A-fragment row / B-fragment column
    const int koff = (lane >> 4) * 2;  // K sub-offset per WMMA layout
    const int node_base = blockIdx.x * 16;

    if (t < 16) rowsq[t] = 0.f;

    // stage the 16x128 activation tile(s)
    {
        int r = t >> 4, c0 = (t & 15) * 8;
        const size_t g = (size_t)(node_base + r) * D + c0;
        cp16(&sA[r * AP + c0],     &H[g]);
        cp16(&sA[r * AP + c0 + 4], &H[g + 4]);
        if (HAS_PROP) {
            cp16(&sP[r * AP + c0],     &P[g]);
            cp16(&sP[r * AP + c0 + 4], &P[g + 4]);
        }
    }

    v8f acc0 = {}, acc1 = {};
    for (int kc = 0; kc < D; kc += 32) {
        cp_wait();        // our async LDS writes (tiles / previous slab) done
        __syncthreads();  // everyone done reading previous slab
        // stage 128x32 weight slab(s), pitch WP
#pragma unroll
        for (int rep = 0; rep < 4; ++rep) {
            int o = (t >> 3) + rep * 32;
            int c = (t & 7) * 4;
            cp16(&sWl[o * WP + c], &Wl[(size_t)o * D + kc + c]);
            if (HAS_PROP)
                cp16(&sWr[o * WP + c], &Wr[(size_t)o * D + kc + c]);
        }
        cp_wait();
        __syncthreads();
#pragma unroll
        for (int k2 = 0; k2 < 32; k2 += 4) {
            v2f a = *(const v2f*)&sA[arow * AP + kc + k2 + koff];
            v2f b = *(const v2f*)&sWl[(wave * 16 + arow) * WP + k2 + koff];
            if (HAS_PROP) {
                acc0 = __builtin_amdgcn_wmma_f32_16x16x4_f32(false, a, false, b,
                                                             (short)0, acc0, false, false);
                v2f ap = *(const v2f*)&sP[arow * AP + kc + k2 + koff];
                v2f bp = *(const v2f*)&sWr[(wave * 16 + arow) * WP + k2 + koff];
                acc1 = __builtin_amdgcn_wmma_f32_16x16x4_f32(false, ap, false, bp,
                                                             (short)0, acc1, false, false);
            } else {
                // split K-chunks across two accumulator chains for ILP
                if (kc & 32)
                    acc1 = __builtin_amdgcn_wmma_f32_16x16x4_f32(false, a, false, b,
                                                                 (short)0, acc1, false, false);
                else
                    acc0 = __builtin_amdgcn_wmma_f32_16x16x4_f32(false, a, false, b,
                                                                 (short)0, acc0, false, false);
            }
        }
    }
    v8f acc = acc0 + acc1;

    // bias (per output column; one column per lane)
    const int colg = wave * 16 + arow;
    float bias = bl[colg];
    if (HAS_PROP) bias += br[colg];
#pragma unroll
    for (int r = 0; r < 8; ++r) acc[r] += bias;

    if (DO_NORM) {
        // per-row sum of squares: reduce this wave's 16 columns, then LDS-combine 8 waves
        float s[8];
#pragma unroll
        for (int r = 0; r < 8; ++r) s[r] = acc[r] * acc[r];
#pragma unroll
        for (int d = 8; d >= 1; d >>= 1)
#pragma unroll
            for (int r = 0; r < 8; ++r) s[r] += __shfl_xor(s[r], d, 32);
        if (lane == 0)
#pragma unroll
            for (int r = 0; r < 8; ++r) atomicAdd(&rowsq[r], s[r]);
        if (lane == 16)
#pragma unroll
            for (int r = 0; r < 8; ++r) atomicAdd(&rowsq[8 + r], s[r]);
        __syncthreads();
#pragma unroll
        for (int r = 0; r < 8; ++r) {
            int row = (lane < 16) ? r : r + 8;
            float inv = 1.0f / fmaxf(sqrtf(rowsq[row]), 1e-12f);
            acc[r] = fmaxf(acc[r] * inv, 0.0f);  // normalize + ReLU
        }
    }

#pragma unroll
    for (int r = 0; r < 8; ++r) {
        int row = (lane < 16) ? r : r + 8;
        out[(size_t)(node_base + row) * D + colg] = acc[r];
    }
}

// ---------------------------------------------------------------------------
// logits = h2 @ W_post2^T + b_post2 (40 cols, padded to 48), then log_softmax.
// 96 threads = 3 waves, each wave one 16x16 column tile of the 16x48 logits.
// ---------------------------------------------------------------------------
__global__ __launch_bounds__(96) void post2_softmax_kernel(
    const float* __restrict__ H2, const float* __restrict__ W2,
    const float* __restrict__ b2, float* __restrict__ out) {
    __shared__ float sA[16 * AP];
    __shared__ float sW[48 * AP];
    __shared__ float pmax[3 * 16];
    __shared__ float psum[3 * 16];

    const int t = threadIdx.x;
    const int lane = t & 31;
    const int wave = t >> 5;  // 0..2
    const int arow = lane & 15;
    const int koff = (lane >> 4) * 2;
    const int node_base = blockIdx.x * 16;

    // stage activation tile (512 float4's over 96 threads)
    for (int i4 = t; i4 < 16 * (D / 4); i4 += 96) {
        int r = i4 >> 5, c = (i4 & 31) * 4;  // 32 float4 per row
        cp16(&sA[r * AP + c], &H2[(size_t)(node_base + r) * D + c]);
    }
    // stage W2 (rows >= 40 zero-filled)
    {
        int o = t >> 1, ch = (t & 1) * 64;
#pragma unroll
        for (int j = 0; j < 64; j += 4) {
            if (o < D_OUT) {
                cp16(&sW[o * AP + ch + j], &W2[(size_t)o * D + ch + j]);
            } else {
                *(float4*)&sW[o * AP + ch + j] = make_float4(0.f, 0.f, 0.f, 0.f);
            }
        }
    }
    cp_wait();
    __syncthreads();

    v8f acc0 = {}, acc1 = {};
#pragma unroll
    for (int k = 0; k < D; k += 8) {  // two independent WMMA chains
        v2f a0 = *(const v2f*)&sA[arow * AP + k + koff];
        v2f b0 = *(const v2f*)&sW[(wave * 16 + arow) * AP + k + koff];
        acc0 = __builtin_amdgcn_wmma_f32_16x16x4_f32(false, a0, false, b0,
                                                     (short)0, acc0, false, false);
        v2f a1 = *(const v2f*)&sA[arow * AP + k + 4 + koff];
        v2f b1 = *(const v2f*)&sW[(wave * 16 + arow) * AP + k + 4 + koff];
        acc1 = __builtin_amdgcn_wmma_f32_16x16x4_f32(false, a1, false, b1,
                                                     (short)0, acc1, false, false);
    }
    v8f acc = acc0 + acc1;

    const int colg = wave * 16 + arow;
    const bool valid = (colg < D_OUT);
    float bias = valid ? b2[colg] : 0.f;
#pragma unroll
    for (int r = 0; r < 8; ++r) acc[r] += bias;

    // row max over 48 lanes-worth of columns (invalid -> -inf)
    float m[8];
#pragma unroll
    for (int r = 0; r < 8; ++r) m[r] = valid ? acc[r] : -3.4e38f;
#pragma unroll
    for (int d = 8; d >= 1; d >>= 1)
#pragma unroll
        for (int r = 0; r < 8; ++r) m[r] = fmaxf(m[r], __shfl_xor(m[r], d, 32));
    if (lane == 0)
#pragma unroll
        for (int r = 0; r < 8; ++r) pmax[wave * 16 + r] = m[r];
    if (lane == 16)
#pragma unroll
        for (int r = 0; r < 8; ++r) pmax[wave * 16 + 8 + r] = m[r];
    __syncthreads();

    float rmax[8];
#pragma unroll
    for (int r = 0; r < 8; ++r) {
        int row = (lane < 16) ? r : r + 8;
        rmax[r] = fmaxf(fmaxf(pmax[row], pmax[16 + row]), pmax[32 + row]);
    }

    float s[8];
#pragma unroll
    for (int r = 0; r < 8; ++r) s[r] = valid ? expf(acc[r] - rmax[r]) : 0.f;
#pragma unroll
    for (int d = 8; d >= 1; d >>= 1)
#pragma unroll
        for (int r = 0; r < 8; ++r) s[r] += __shfl_xor(s[r], d, 32);
    if (lane == 0)
#pragma unroll
        for (int r = 0; r < 8; ++r) psum[wave * 16 + r] = s[r];
    if (lane == 16)
#pragma unroll
        for (int r = 0; r < 8; ++r) psum[wave * 16 + 8 + r] = s[r];
    __syncthreads();

    if (valid) {
#pragma unroll
        for (int r = 0; r < 8; ++r) {
            int row = (lane < 16) ? r : r + 8;
            float tot = psum[row] + psum[16 + row] + psum[32 + row];
            out[(size_t)(node_base + row) * D_OUT + colg] =
                acc[r] - rmax[r] - logf(tot);
        }
    }
}

// ---------------------------------------------------------------------------
extern "C" void kernel_launch(void* const* d_in, const int* in_sizes, int n_in,
                              void* d_out, int out_size, void* d_ws, size_t ws_size,
                              hipStream_t stream) {
    const float*      x   = (const float*)d_in[0];
    const long long*  ei  = (const long long*)d_in[1];
    const float*      Wl  = (const float*)d_in[2];
    const float*      bl  = (const float*)d_in[3];
    const float*      Wr  = (const float*)d_in[4];
    const float*      br  = (const float*)d_in[5];
    const float*      Wp1 = (const float*)d_in[6];
    const float*      bp1 = (const float*)d_in[7];
    const float*      Wp2 = (const float*)d_in[8];
    const float*      bp2 = (const float*)d_in[9];
    float*            out = (float*)d_out;

    const size_t nfeat = (size_t)N_NODES * D;  // 12.8M floats
    float* prop = (float*)d_ws;
    float* h0   = prop + nfeat;
    float* h1   = h0 + nfeat;

    const int tiles = N_NODES / 16;  // 6250
    const float* hcur = x;
    float* bufs[2] = {h0, h1};

    for (int L = 0; L < 3; ++L) {
        zero_kernel<<<(int)(nfeat / 4 / 256), 256, 0, stream>>>((float4*)prop,
                                                                (int)(nfeat / 4));
        scatter_kernel<<<N_EDGES / 8, 256, 0, stream>>>(hcur, ei, prop);
        float* hn = bufs[L & 1];
        sage_gemm_kernel<1, 1><<<tiles, 256, 0, stream>>>(
            hcur, prop, Wl + (size_t)L * D * D, bl + (size_t)L * D,
            Wr + (size_t)L * D * D, br + (size_t)L * D, hn);
        hcur = hn;
    }

    float* h2 = (hcur == h0) ? h1 : h0;
    sage_gemm_kernel<0, 0><<<tiles, 256, 0, stream>>>(
        hcur, nullptr, Wp1, bp1, nullptr, nullptr, h2);
    post2_softmax_kernel<<<tiles, 96, 0, stream>>>(h2, Wp2, bp2, out);
}